// SpatialAttentionLoss_25649544691746
// MI455X (gfx1250) — compile-verified
//
#include <hip/hip_runtime.h>

typedef float v2f __attribute__((ext_vector_type(2)));
typedef float v8f __attribute__((ext_vector_type(8)));

#define HW      262144          // 512*512
#define NS      96              // B*S = 8*12
#define PCHUNK  64              // chunks per slice
#define CHUNK   4096            // HW / PCHUNK
#define THREADS 256
#define EPSV    1e-6f

// ---------------------------------------------------------------------------
// Wave(32) reduction using V_WMMA_F32_16X16X4_F32:
//   A = ones(16x4), B = {acc_vgpr, 0}  =>  D[m,n] = acc[n] + acc[n+16]  (all m)
// D's first VGPR then holds s[l%16] in every lane; 4 xor-shuffles finish.
// EXEC is all-ones here (full block, no divergence) as WMMA requires.
// ---------------------------------------------------------------------------
__device__ __forceinline__ float wave_reduce_wmma(float v) {
    v2f a; a[0] = 1.0f; a[1] = 1.0f;       // all-ones A (16x4 f32, 2 VGPRs)
    v2f b; b[0] = v;    b[1] = 0.0f;       // lane accumulators as B, rest zero
    v8f c = {};
    c = __builtin_amdgcn_wmma_f32_16x16x4_f32(
            /*neg_a=*/false, a, /*neg_b=*/false, b,
            /*c_mod=*/(short)0, c, /*reuse_a=*/false, /*reuse_b=*/false);
    float s = c[0];                        // s = v[l%16] + v[l%16 + 16]
    s += __shfl_xor(s, 8, 32);
    s += __shfl_xor(s, 4, 32);
    s += __shfl_xor(s, 2, 32);
    s += __shfl_xor(s, 1, 32);
    return s;                              // total over all 32 lanes
}

// Per-element math: BCE-with-logits, dice sums, attention BCE.
__device__ __forceinline__ void accum_elem(float xx, float pp, float tt, float aa,
                                           float& sb, float& si, float& sp,
                                           float& st, float& sa) {
    // stable BCE with logits: max(x,0) - x*t + log(1+exp(-|x|))
    float e = __expf(-fabsf(xx));
    sb += fmaxf(xx, 0.0f) - xx * tt + __logf(1.0f + e);
    // dice numerator/denominator sums; fp-suppression shares sp (p>0)
    si = fmaf(pp, tt, si);
    sp += fmaxf(pp, 0.0f);
    st += tt;
    // attention BCE vs binarized target (targets are exactly 0/1)
    float tb  = (tt > 0.5f) ? 1.0f : 0.0f;
    float la  = fmaxf(__logf(aa), -100.0f);
    float l1a = fmaxf(__logf(1.0f - aa), -100.0f);
    sa -= tb * la + (1.0f - tb) * l1a;
}

__global__ __launch_bounds__(THREADS)
void loss_partial_kernel(const float* __restrict__ x,   // seg_logits
                         const float* __restrict__ p,   // seg_probs
                         const float* __restrict__ t,   // seg_targets
                         const float* __restrict__ a,   // attention_maps
                         float* __restrict__ partials)  // [NS*PCHUNK][5]
{
    const int    blk  = blockIdx.x;                 // = slice*PCHUNK + chunk
    const size_t base = (size_t)blk * CHUNK;        // slices are contiguous
    const int    tid  = threadIdx.x;

    const float4* x4 = (const float4*)(x + base);
    const float4* p4 = (const float4*)(p + base);
    const float4* t4 = (const float4*)(t + base);
    const float4* a4 = (const float4*)(a + base);

    float sb = 0.f, si = 0.f, sp = 0.f, st = 0.f, sa = 0.f;

#pragma unroll
    for (int i = 0; i < CHUNK / (THREADS * 4); ++i) {
        const int v = i * THREADS + tid;            // coalesced b128 loads
        float4 xv = x4[v];
        float4 pv = p4[v];
        float4 tv = t4[v];
        float4 av = a4[v];
        accum_elem(xv.x, pv.x, tv.x, av.x, sb, si, sp, st, sa);
        accum_elem(xv.y, pv.y, tv.y, av.y, sb, si, sp, st, sa);
        accum_elem(xv.z, pv.z, tv.z, av.z, sb, si, sp, st, sa);
        accum_elem(xv.w, pv.w, tv.w, av.w, sb, si, sp, st, sa);
    }

    float r[5] = { sb, si, sp, st, sa };
#pragma unroll
    for (int j = 0; j < 5; ++j) r[j] = wave_reduce_wmma(r[j]);

    __shared__ float smem[THREADS / 32][5];
    const int wave = tid >> 5, lane = tid & 31;
    if (lane == 0) {
#pragma unroll
        for (int j = 0; j < 5; ++j) smem[wave][j] = r[j];
    }
    __syncthreads();
    if (tid < 5) {
        float acc = 0.f;
#pragma unroll
        for (int w = 0; w < THREADS / 32; ++w) acc += smem[w][tid];
        partials[(size_t)blk * 5 + tid] = acc;
    }
}

__global__ __launch_bounds__(128)
void loss_final_kernel(const float* __restrict__ partials,        // [NS*PCHUNK][5]
                       const float* __restrict__ presence_probs,  // [NS]
                       const float* __restrict__ presence_targets,// [NS]
                       float* __restrict__ out)                   // [7]
{
    const int tid = threadIdx.x;
    float v[7] = {0.f, 0.f, 0.f, 0.f, 0.f, 0.f, 0.f};

    if (tid < NS) {
        float sb = 0.f, si = 0.f, sp = 0.f, st = 0.f, sa = 0.f;
        for (int c = 0; c < PCHUNK; ++c) {
            const float* q = partials + ((size_t)tid * PCHUNK + c) * 5;
            sb += q[0]; si += q[1]; sp += q[2]; st += q[3]; sa += q[4];
        }
        const float pt = presence_targets[tid];
        const float pp = presence_probs[tid];
        const float m  = (pt != 0.0f) ? 1.0f : 0.0f;
        const float inv = 1.0f / (float)HW;

        const float segbce = sb * inv;
        const float dice   = 1.0f - (2.0f * si + EPSV) / (sp + st + EPSV);
        const float fp     = sp * inv;
        const float att    = sa * inv;
        // presence BCE (accurate libm on tiny array)
        const float absb = -(pt * fmaxf(logf(pp), -100.0f)
                           + (1.0f - pt) * fmaxf(log1pf(-pp), -100.0f));
        float conf = 0.0f;
        if (pt == 0.0f && pp > 0.5f) conf = (pp - 0.5f) * (pp - 0.5f);
        if (pt == 1.0f && pp < 0.5f) conf = (0.5f - pp) * (0.5f - pp);

        v[0] = m;                  // n_valid
        v[1] = segbce * m;
        v[2] = dice * m;
        v[3] = fp * (1.0f - m);
        v[4] = att * m;
        v[5] = absb;
        v[6] = conf;
    }

    __shared__ float red[128][7];
#pragma unroll
    for (int j = 0; j < 7; ++j) red[tid][j] = v[j];
    __syncthreads();
    for (int s = 64; s > 0; s >>= 1) {
        if (tid < s) {
#pragma unroll
            for (int j = 0; j < 7; ++j) red[tid][j] += red[tid + s][j];
        }
        __syncthreads();
    }

    if (tid == 0) {
        const float n_valid  = red[0][0];
        const float n_absent = (float)NS - n_valid;
        const float seg_loss  = (n_valid  > 0.f) ? red[0][1] / fmaxf(n_valid, 1.f)  : 0.f;
        const float dice_loss = (n_valid  > 0.f) ? red[0][2] / fmaxf(n_valid, 1.f)  : 0.f;
        const float fp_loss   = (n_absent > 0.f) ? red[0][3] / fmaxf(n_absent, 1.f) : 0.f;
        const float att_loss     = red[0][4] / (float)NS;
        const float absence_loss = red[0][5] / (float)NS;
        const float conf_loss    = red[0][6] / (float)NS;
        const float total = 1.0f * seg_loss + 1.0f * dice_loss
                          + 2.0f * absence_loss + 0.3f * att_loss
                          + 0.1f * conf_loss + 0.5f * fp_loss;
        out[0] = total;
        out[1] = seg_loss;
        out[2] = dice_loss;
        out[3] = absence_loss;
        out[4] = att_loss;
        out[5] = conf_loss;
        out[6] = fp_loss;
    }
}

extern "C" void kernel_launch(void* const* d_in, const int* in_sizes, int n_in,
                              void* d_out, int out_size, void* d_ws, size_t ws_size,
                              hipStream_t stream) {
    (void)in_sizes; (void)n_in; (void)out_size; (void)ws_size;
    // setup_inputs() dict order:
    const float* seg_logits       = (const float*)d_in[0];
    const float* seg_probs        = (const float*)d_in[1];
    const float* presence_probs   = (const float*)d_in[2];
    const float* attention_maps   = (const float*)d_in[3];
    const float* seg_targets      = (const float*)d_in[4];
    const float* presence_targets = (const float*)d_in[5];
    float* out      = (float*)d_out;
    float* partials = (float*)d_ws;   // NS*PCHUNK*5*4 = 122,880 bytes

    loss_partial_kernel<<<NS * PCHUNK, THREADS, 0, stream>>>(
        seg_logits, seg_probs, seg_targets, attention_maps, partials);
    loss_final_kernel<<<1, 128, 0, stream>>>(
        partials, presence_probs, presence_targets, out);
}